// FOFE_Encoding_6287832121915
// MI455X (gfx1250) — compile-verified
//
#include <hip/hip_runtime.h>
#include <hip/hip_bf16.h>
#include <stdint.h>

#define VOCAB            512
#define WRD              24
#define B_DIM            64
#define S_DIM            512
#define ROWS_TOTAL       (B_DIM * S_DIM)      // 32768 rows of 512 f32
#define WAVES_PER_BLOCK  8                    // 256 threads = 8 wave32

// One wave32 per (b,s) row:
//  - lane k<24 holds char c_k, weight w_k = (c_k!=0) * alpha^(#nonpad after k)
//  - accumulate into a wave-private 512-float LDS row with ds_add_f32
//  - stream the row to global via gfx1250 async LDS->global movers
__global__ __launch_bounds__(256)
void fofe_encoding_kernel(const int* __restrict__ sents,
                          const int* __restrict__ lengths,
                          const float* __restrict__ alpha_p,
                          float* __restrict__ out)
{
    __shared__ float smem[WAVES_PER_BLOCK][VOCAB];   // 16 KB per block

    const int tid  = threadIdx.x;
    const int wave = tid >> 5;
    const int lane = tid & 31;
    const int row  = blockIdx.x * WAVES_PER_BLOCK + wave;   // 0..32767

    // Tail of d_out: lengths cast to the f32 output buffer.
    if (blockIdx.x == 0 && tid < B_DIM) {
        out[(size_t)ROWS_TOTAL * VOCAB + tid] = (float)lengths[tid];
    }

    float* rowp = &smem[wave][0];

    // ---- zero the wave-private row: 4x ds_store_b128 per lane (16 floats) ----
    float4 z = make_float4(0.f, 0.f, 0.f, 0.f);
    float4* rowv = (float4*)rowp;
#pragma unroll
    for (int i = 0; i < 4; ++i)
        rowv[lane + i * 32] = z;

    // ---- per-lane char + FOFE weight ----
    int c = 0;
    if (lane < WRD) c = sents[row * WRD + lane];
    const bool m = (c != 0);

    unsigned mask   = __builtin_amdgcn_ballot_w32(m);       // wave32 ballot
    int      suffix = __popc(mask & (0xFFFFFFFEu << lane)); // nonpad strictly after k

    const float alpha = alpha_p[0];                          // uniform scalar
    if (m) {
        // alpha^suffix = exp2(suffix * log2(alpha)); native v_log_f32/v_exp_f32
        float w = __builtin_amdgcn_exp2f((float)suffix * __builtin_amdgcn_logf(alpha));
        atomicAdd(rowp + c, w);                              // ds_add_f32 (no rtn)
    }

    // All DS work (zeroing stores + atomics) from this wave must land before
    // the async engine reads LDS (ASYNC ops are NOT ordered with DS ops).
    asm volatile("s_wait_dscnt 0x0" ::: "memory");

    // ---- stream the 2KB row to global memory: async LDS -> global ----
    float* orow = out + (size_t)row * VOCAB;
#pragma unroll
    for (int i = 0; i < 4; ++i) {
        const float* g     = orow + lane * 4 + i * 128;                    // 16B per lane
        unsigned     laddr = (unsigned)(uintptr_t)rowp + (unsigned)(lane * 16 + i * 512);
        asm volatile("global_store_async_from_lds_b128 %0, %1, off"
                     :: "v"(g), "v"(laddr)
                     : "memory");
    }
    // Row buffer is wave-private and not reused, but retire ASYNCcnt cleanly.
    asm volatile("s_wait_asynccnt 0x0" ::: "memory");
}

extern "C" void kernel_launch(void* const* d_in, const int* in_sizes, int n_in,
                              void* d_out, int out_size, void* d_ws, size_t ws_size,
                              hipStream_t stream) {
    const int*   sents   = (const int*)d_in[0];   // [64,512,24] int
    const int*   lengths = (const int*)d_in[1];   // [64] int
    const float* alpha   = (const float*)d_in[2]; // [1] f32
    float*       out     = (float*)d_out;         // [64*512*512] f32 ++ [64] lengths

    dim3 grid(ROWS_TOTAL / WAVES_PER_BLOCK);      // 4096 blocks
    dim3 block(256);                              // 8 wave32 per block
    fofe_encoding_kernel<<<grid, block, 0, stream>>>(sents, lengths, alpha, out);
}